// LogEig_89756226552161
// MI455X (gfx1250) — compile-verified
//
#include <hip/hip_runtime.h>
#include <math.h>

// ---------------------------------------------------------------------------
// Batched SPD matrix logarithm, 64x64 fp32, via inverse scaling-and-squaring:
//   Y = P/||P||_F  (eig in (0,1])
//   6x repeated Newton-Schulz matrix square roots (matmul-only, coupled form)
//   log of near-identity M via 8-term Mercator series (Horner)
//   log(P) = 64*log(M) + log(s)*I
// One workgroup (8 wave32) per matrix; all matmuls are v_wmma_f32_16x16x4_f32.
// Each wave computes two C tiles in the SAME tile-row so the A fragment is
// loaded once per K-step and shared by both WMMAs (2 accumulators).
// ---------------------------------------------------------------------------

#define LDW 68                       // LDS row stride (floats): conflict-free, 16B-aligned rows
#define BUF_FLOATS (64 * LDW)        // one 64x64 padded matrix
#define NSQRT 6
#define SQ_SCALE 64.0f               // 2^NSQRT

typedef __attribute__((ext_vector_type(2))) float v2f;
typedef __attribute__((ext_vector_type(8))) float v8f;

// 64x64 * 64x64 matmul entirely in LDS. 16 output tiles of 16x16; wave w owns
// tiles (ti = w>>1, tj = 2*(w&1)) and (ti, tj+1): shared A fragment, 2 accs.
__device__ __forceinline__ void mm64(const float* __restrict__ A,
                                     const float* __restrict__ B,
                                     float* __restrict__ C,
                                     int lane, int wave)
{
    const int m   = lane & 15;       // row within tile (A) / col within tile (B,C)
    const int kh  = lane >> 4;       // K half-select per ISA 16x4 A layout
    const int ti  = wave >> 1;       // C tile row
    const int tj0 = (wave & 1) * 2;  // first of two adjacent C tile cols

    v8f acc0 = {0.f, 0.f, 0.f, 0.f, 0.f, 0.f, 0.f, 0.f};
    v8f acc1 = {0.f, 0.f, 0.f, 0.f, 0.f, 0.f, 0.f, 0.f};

    const float* Ar = A + (ti * 16 + m) * LDW;   // A row (shared by both tiles)
    const float* B0 = B + tj0 * 16 + m;          // column n = lane&15, tile tj0
    const float* B1 = B0 + 16;                   // column n, tile tj0+1

#pragma unroll
    for (int k = 0; k < 64; k += 4) {
        const int ka = k + 2 * kh;
        v2f a, b0, b1;
        a.x  = Ar[ka];                 // A[m][ka]
        a.y  = Ar[ka + 1];             // A[m][ka+1]
        b0.x = B0[ka * LDW];           // B[ka][n]       (tile tj0)
        b0.y = B0[(ka + 1) * LDW];     // B[ka+1][n]
        b1.x = B1[ka * LDW];           // B[ka][n+16]    (tile tj0+1)
        b1.y = B1[(ka + 1) * LDW];     // B[ka+1][n+16]
        acc0 = __builtin_amdgcn_wmma_f32_16x16x4_f32(
            false, a, false, b0, (short)0, acc0, false, false);
        acc1 = __builtin_amdgcn_wmma_f32_16x16x4_f32(
            false, a, false, b1, (short)0, acc1, false, false);
    }

    // C/D layout: VGPR v -> row (v + 8*kh), col n = lane&15
    const int mb = ti * 16 + kh * 8;
    float* C0 = C + tj0 * 16 + m;
    float* C1 = C0 + 16;
#pragma unroll
    for (int v = 0; v < 8; ++v) {
        C0[(mb + v) * LDW] = acc0[v];
        C1[(mb + v) * LDW] = acc1[v];
    }
}

__global__ __launch_bounds__(256)
void logm64_kernel(const float* __restrict__ P, float* __restrict__ Out)
{
    extern __shared__ float buf[];       // 5 padded 64x64 buffers (87 KB)
    __shared__ float s_norm2;

    float* Y = buf + 0 * BUF_FLOATS;
    float* Z = buf + 1 * BUF_FLOATS;
    float* T = buf + 2 * BUF_FLOATS;
    float* U = buf + 3 * BUF_FLOATS;
    float* V = buf + 4 * BUF_FLOATS;

    const int tid  = threadIdx.x;
    const int lane = tid & 31;
    const int wave = tid >> 5;

    const float* __restrict__ Pm = P   + (size_t)blockIdx.x * 4096;
    float*       __restrict__ Om = Out + (size_t)blockIdx.x * 4096;

    if (tid == 0) s_norm2 = 0.0f;
    __syncthreads();

    // ---- Load P -> Y (float4, rows 16B aligned since LDW*4 = 272) and reduce ||P||_F^2
    float part = 0.0f;
#pragma unroll
    for (int i = 0; i < 4; ++i) {
        const int e4 = tid + i * 256;            // float4 index 0..1023
        const int e  = e4 * 4;
        const int r  = e >> 6, c = e & 63;
        const float4 v = ((const float4*)Pm)[e4];
        *(float4*)&Y[r * LDW + c] = v;
        part += v.x * v.x + v.y * v.y + v.z * v.z + v.w * v.w;
    }
#pragma unroll
    for (int off = 16; off > 0; off >>= 1)
        part += __shfl_down(part, off, 32);
    if (lane == 0) atomicAdd(&s_norm2, part);    // ds_add_f32
    __syncthreads();

    const float s     = sqrtf(s_norm2);          // >= lambda_max(P)
    const float inv_s = 1.0f / s;

    // ---- Y = P/s
#pragma unroll
    for (int i = 0; i < 16; ++i) {
        const int e = tid + i * 256;
        const int r = e >> 6, c = e & 63;
        Y[r * LDW + c] *= inv_s;
    }
    __syncthreads();

    // ---- NSQRT repeated square roots via coupled Newton-Schulz
    const int iters[NSQRT] = {10, 8, 7, 6, 5, 4};
#pragma unroll 1
    for (int sq = 0; sq < NSQRT; ++sq) {
        // Z = I
#pragma unroll
        for (int i = 0; i < 16; ++i) {
            const int e = tid + i * 256;
            const int r = e >> 6, c = e & 63;
            Z[r * LDW + c] = (r == c) ? 1.0f : 0.0f;
        }
        __syncthreads();

#pragma unroll 1
        for (int it = 0; it < iters[sq]; ++it) {
            mm64(Z, Y, T, lane, wave);           // T = Z*Y
            __syncthreads();
#pragma unroll
            for (int i = 0; i < 16; ++i) {       // T := 1.5I - 0.5T  (= W)
                const int e = tid + i * 256;
                const int r = e >> 6, c = e & 63;
                float x = -0.5f * T[r * LDW + c];
                if (r == c) x += 1.5f;
                T[r * LDW + c] = x;
            }
            __syncthreads();
            mm64(Y, T, U, lane, wave);           // Ynew = Y*W
            mm64(T, Z, V, lane, wave);           // Znew = W*Z
            __syncthreads();
            float* tmp;
            tmp = Y; Y = U; U = tmp;
            tmp = Z; Z = V; V = tmp;
        }
        // Y ~= sqrt of previous Y
    }

    // ---- log of near-identity M (=Y): X = I - M (into Z), Horner 8-term series into U
#pragma unroll
    for (int i = 0; i < 16; ++i) {
        const int e = tid + i * 256;
        const int r = e >> 6, c = e & 63;
        float x = -Y[r * LDW + c];
        if (r == c) x += 1.0f;
        Z[r * LDW + c] = x;                      // X
    }
    __syncthreads();
#pragma unroll
    for (int i = 0; i < 16; ++i) {
        const int e = tid + i * 256;
        const int r = e >> 6, c = e & 63;
        U[r * LDW + c] = 0.125f * Z[r * LDW + c];   // R = X/8
    }
    __syncthreads();
#pragma unroll 1
    for (int j = 7; j >= 1; --j) {
        mm64(Z, U, T, lane, wave);               // T = X*R
        __syncthreads();
        const float cj = 1.0f / (float)j;
#pragma unroll
        for (int i = 0; i < 16; ++i) {
            const int e = tid + i * 256;
            const int r = e >> 6, c = e & 63;
            U[r * LDW + c] = T[r * LDW + c] + cj * Z[r * LDW + c];
        }
        __syncthreads();
    }
    // now log(M) = -U ;  log(P) = -SQ_SCALE*U + log(s)*I

    const float lgs = logf(s);
#pragma unroll
    for (int i = 0; i < 16; ++i) {
        const int e = tid + i * 256;
        const int r = e >> 6, c = e & 63;
        float x = -SQ_SCALE * U[r * LDW + c];
        if (r == c) x += lgs;
        Om[e] = x;                               // coalesced
    }
}

extern "C" void kernel_launch(void* const* d_in, const int* in_sizes, int n_in,
                              void* d_out, int out_size, void* d_ws, size_t ws_size,
                              hipStream_t stream)
{
    const float* P = (const float*)d_in[0];
    float* Out = (float*)d_out;
    const int nmat = in_sizes[0] / 4096;         // B*H = 8192 matrices of 64x64
    const size_t shmem = (size_t)5 * BUF_FLOATS * sizeof(float);  // 87,040 B
    logm64_kernel<<<dim3(nmat), dim3(256), shmem, stream>>>(P, Out);
}